// GATLayer_498216206816
// MI455X (gfx1250) — compile-verified
//
#include <hip/hip_runtime.h>

typedef float v2f __attribute__((ext_vector_type(2)));
typedef float v8f __attribute__((ext_vector_type(8)));

#define FIN   128
#define NHEAD 8
#define FOUT  16
#define FTOT  128   // NHEAD * FOUT
#define ELU_ALPHA 0.2f

// ---------------------------------------------------------------------------
// Kernel 1: feats = x @ W  via V_WMMA_F32_16X16X4_F32.
// One wave per 16x16 output tile. Block = 256 threads = 8 waves; wave w covers
// output columns [16w, 16w+16) (i.e. one head). Grid.x covers 16-row stripes.
// ---------------------------------------------------------------------------
__global__ void gat_gemm_feats(const float* __restrict__ x,
                               const float* __restrict__ W,
                               float* __restrict__ feats, int N) {
    const int wave = threadIdx.x >> 5;          // 0..7 -> column tile / head
    const int lane = threadIdx.x & 31;
    const int row0 = blockIdx.x * 16;
    if (row0 >= N) return;
    const int col0 = wave * 16;
    const int m    = lane & 15;                 // row (A) / col (B,C) index
    const int half = lane >> 4;                 // K-pair select for A/B frags
    const int ksub = half * 2;

    v8f c = {0.f, 0.f, 0.f, 0.f, 0.f, 0.f, 0.f, 0.f};
    const float* arow = x + (size_t)(row0 + m) * FIN;

#pragma unroll
    for (int k = 0; k < FIN; k += 4) {
        v2f a, b;
        // A 16x4 f32: VGPR0 = K=0 (lanes 0-15) / K=2 (lanes 16-31); VGPR1 = K=1 / K=3
        a.x = arow[k + ksub];
        a.y = arow[k + ksub + 1];
        // B 4x16 f32: row K striped across lanes (symmetric K-pair split)
        b.x = W[(size_t)(k + ksub) * FTOT + col0 + m];
        b.y = W[(size_t)(k + ksub + 1) * FTOT + col0 + m];
        c = __builtin_amdgcn_wmma_f32_16x16x4_f32(
                /*neg_a=*/false, a, /*neg_b=*/false, b,
                /*c_mod=*/(short)0, c, /*reuse_a=*/false, /*reuse_b=*/false);
    }

    // C/D 16x16 f32: VGPR r -> M = r (lanes 0-15) / M = r+8 (lanes 16-31), N = lane&15
#pragma unroll
    for (int r = 0; r < 8; ++r) {
        const int row = row0 + r + half * 8;
        feats[(size_t)row * FTOT + col0 + m] = c[r];
    }
}

// ---------------------------------------------------------------------------
// Kernel 2: per-(node, head) attention scores (two 16-wide dot products).
// Scores stored node-major: score[n*8 + h].
// ---------------------------------------------------------------------------
__global__ void gat_scores(const float* __restrict__ feats,
                           const float* __restrict__ ssc,   // (8,1,16)
                           const float* __restrict__ tsc,   // (8,1,16)
                           float* __restrict__ src_score,
                           float* __restrict__ tgt_score, int N) {
    const int i = blockIdx.x * blockDim.x + threadIdx.x;
    if (i >= N * NHEAD) return;
    const int n = i >> 3;
    const int h = i & 7;
    const float* f = feats + (size_t)n * FTOT + h * FOUT;
    const float* s = ssc + h * FOUT;
    const float* t = tsc + h * FOUT;
    float as = 0.f, at = 0.f;
#pragma unroll
    for (int j = 0; j < FOUT; ++j) {
        const float v = f[j];
        as += v * s[j];
        at += v * t[j];
    }
    src_score[i] = as;
    tgt_score[i] = at;
}

// ---------------------------------------------------------------------------
// Zero-fill helper (d_out / denom are poisoned by the harness).
// ---------------------------------------------------------------------------
__global__ void gat_zero(float* __restrict__ p, long long n) {
    const long long i = (long long)blockIdx.x * blockDim.x + threadIdx.x;
    if (i < n) p[i] = 0.f;
}

__device__ __forceinline__ float elu_f(float v) {
    return v > 0.f ? v : ELU_ALPHA * (__expf(v) - 1.f);
}

// ---------------------------------------------------------------------------
// Kernel 3 (edge pass 1): denom[tgt][h] += exp(elu(s_src + s_tgt)).
// One thread per edge, all 8 heads -> contiguous 32B score reads, L2 atomics.
// ---------------------------------------------------------------------------
__global__ void gat_edge_denom(const int* __restrict__ ei,
                               const float* __restrict__ ss,
                               const float* __restrict__ ts,
                               float* __restrict__ denom, int E) {
    const int e = blockIdx.x * blockDim.x + threadIdx.x;
    if (e >= E) return;
    const int src = ei[2 * e];
    const int tgt = ei[2 * e + 1];
#pragma unroll
    for (int h = 0; h < NHEAD; ++h) {
        const float sc = elu_f(ss[(size_t)src * NHEAD + h] + ts[(size_t)tgt * NHEAD + h]);
        atomicAdd(&denom[(size_t)tgt * NHEAD + h], __expf(sc));
    }
}

// ---------------------------------------------------------------------------
// Kernel 4 (edge pass 2): out[tgt][h*16+f] += att * feats[src][h*16+f].
// One thread per (edge, head): 64B gather + 16 f32 L2 atomics.
// ---------------------------------------------------------------------------
__global__ void gat_edge_scatter(const int* __restrict__ ei,
                                 const float* __restrict__ ss,
                                 const float* __restrict__ ts,
                                 const float* __restrict__ denom,
                                 const float* __restrict__ feats,
                                 float* __restrict__ out, int E) {
    const int idx = blockIdx.x * blockDim.x + threadIdx.x;
    if (idx >= E * NHEAD) return;
    const int e = idx >> 3;
    const int h = idx & 7;
    const int src = ei[2 * e];
    const int tgt = ei[2 * e + 1];
    const float sc  = elu_f(ss[(size_t)src * NHEAD + h] + ts[(size_t)tgt * NHEAD + h]);
    const float att = __expf(sc) / denom[(size_t)tgt * NHEAD + h];
    const float* f = feats + (size_t)src * FTOT + h * FOUT;
    float*       o = out   + (size_t)tgt * FTOT + h * FOUT;
#pragma unroll
    for (int j = 0; j < FOUT; ++j) {
        atomicAdd(&o[j], f[j] * att);
    }
}

// ---------------------------------------------------------------------------
extern "C" void kernel_launch(void* const* d_in, const int* in_sizes, int n_in,
                              void* d_out, int out_size, void* d_ws, size_t ws_size,
                              hipStream_t stream) {
    const float* x   = (const float*)d_in[0];   // (N, 128)
    const float* W   = (const float*)d_in[1];   // (128, 128)
    const float* ssc = (const float*)d_in[2];   // (8, 1, 16)
    const float* tsc = (const float*)d_in[3];   // (8, 1, 16)
    const int*   ei  = (const int*)  d_in[4];   // (E, 2)
    float* out = (float*)d_out;                 // (N, 128)

    const int N = in_sizes[0] / FIN;
    const int E = in_sizes[4] / 2;

    // Workspace layout (fp32): feats | src_score | tgt_score | denom
    float* feats = (float*)d_ws;
    float* ss    = feats + (size_t)N * FTOT;
    float* ts    = ss    + (size_t)N * NHEAD;
    float* denom = ts    + (size_t)N * NHEAD;

    // 1) GEMM via WMMA f32 16x16x4
    gat_gemm_feats<<<(N + 15) / 16, 256, 0, stream>>>(x, W, feats, N);

    // 2) per-(node, head) scores
    {
        const int nt = N * NHEAD;
        gat_scores<<<(nt + 255) / 256, 256, 0, stream>>>(feats, ssc, tsc, ss, ts, N);
    }

    // 3) zero denom and out
    {
        const long long nd = (long long)N * NHEAD;
        gat_zero<<<(unsigned)((nd + 255) / 256), 256, 0, stream>>>(denom, nd);
        const long long no = (long long)N * FTOT;
        gat_zero<<<(unsigned)((no + 255) / 256), 256, 0, stream>>>(out, no);
    }

    // 4) edge pass 1: softmax denominators (L2-resident atomics)
    gat_edge_denom<<<(E + 255) / 256, 256, 0, stream>>>(ei, ss, ts, denom, E);

    // 5) edge pass 2: attention-weighted scatter-add
    {
        const int nt = E * NHEAD;
        gat_edge_scatter<<<(nt + 255) / 256, 256, 0, stream>>>(ei, ss, ts, denom, feats, out, E);
    }
}